// NeedlemanWunschAligner_37340445671775
// MI455X (gfx1250) — compile-verified
//
#include <hip/hip_runtime.h>
#include <hip/hip_bf16.h>

typedef __attribute__((ext_vector_type(2))) float v2f;
typedef __attribute__((ext_vector_type(8))) float v8f;

#define NEGC (-1.0e9f)
#define NSEQ 1024
#define DIMK 512
#define NSYM 22

// ---------------------------------------------------------------------------
// Kernel 1: zemb[32][512] = emb_pad[32][512] @ W[512][512] + b   (WMMA f32)
// grid: 64 blocks x 32 threads (1 wave per 16x16 output tile; 2x32 tiles)
// ---------------------------------------------------------------------------
__global__ __launch_bounds__(32) void zemb_wmma_kernel(
    const float* __restrict__ emb, const float* __restrict__ W,
    const float* __restrict__ bias, float* __restrict__ zemb) {
  int tile = blockIdx.x;
  int tm = tile >> 5;          // 0..1  (rows of 16)
  int tn = tile & 31;          // 0..31 (cols of 16)
  int lane = threadIdx.x;
  int l16 = lane & 15;
  int khalf = (lane >> 4) << 1;  // 0 or 2

  int mrow = tm * 16 + l16;      // A row for this lane
  int ncol = tn * 16 + l16;      // B col for this lane

  v8f acc = {};
  for (int k0 = 0; k0 < DIMK; k0 += 4) {
    int ka = k0 + khalf;
    v2f a, bb;
    // A fragment: 16x4 f32; lane l holds A[l%16][2*(l/16)+r], r=0,1
    if (mrow < NSYM) {
      a.x = emb[mrow * DIMK + ka];
      a.y = emb[mrow * DIMK + ka + 1];
    } else {
      a.x = 0.f; a.y = 0.f;
    }
    // B fragment: 4x16 f32; lane l holds B[2*(l/16)+r][l%16]
    bb.x = W[(size_t)ka * DIMK + ncol];
    bb.y = W[(size_t)(ka + 1) * DIMK + ncol];
    acc = __builtin_amdgcn_wmma_f32_16x16x4_f32(
        false, a, false, bb, (short)0, acc, false, false);
  }
  float bv = bias[ncol];
  #pragma unroll
  for (int r = 0; r < 8; ++r) {
    int row = tm * 16 + r + ((lane >> 4) << 3);  // C/D: VGPR r -> M=r / r+8
    zemb[row * DIMK + ncol] = acc[r] + bv;
  }
}

// ---------------------------------------------------------------------------
// Kernel 2: Gram G[32][32] = zemb @ zemb^T   (WMMA f32)
// grid: 4 blocks x 32 threads (2x2 tiles)
// ---------------------------------------------------------------------------
__global__ __launch_bounds__(32) void gram_wmma_kernel(
    const float* __restrict__ zemb, float* __restrict__ G) {
  int tm = blockIdx.x >> 1;
  int tn = blockIdx.x & 1;
  int lane = threadIdx.x;
  int l16 = lane & 15;
  int khalf = (lane >> 4) << 1;

  int arow = tm * 16 + l16;
  int brow = tn * 16 + l16;    // B[k][n] = zemb[n][k]

  v8f acc = {};
  for (int k0 = 0; k0 < DIMK; k0 += 4) {
    int ka = k0 + khalf;
    v2f a, bb;
    a.x = zemb[arow * DIMK + ka];
    a.y = zemb[arow * DIMK + ka + 1];
    bb.x = zemb[brow * DIMK + ka];
    bb.y = zemb[brow * DIMK + ka + 1];
    acc = __builtin_amdgcn_wmma_f32_16x16x4_f32(
        false, a, false, bb, (short)0, acc, false, false);
  }
  #pragma unroll
  for (int r = 0; r < 8; ++r) {
    int row = tm * 16 + r + ((lane >> 4) << 3);
    int col = tn * 16 + l16;
    G[row * 32 + col] = acc[r];
  }
}

// ---------------------------------------------------------------------------
// Kernel 3: u[a] = zemb[a] . Wg[0:512], v[a] = zemb[a] . Wg[512:1024]
// grid: 22 blocks x 256 threads
// ---------------------------------------------------------------------------
__global__ __launch_bounds__(256) void uv_kernel(
    const float* __restrict__ zemb, const float* __restrict__ Wg,
    float* __restrict__ uv) {
  int a = blockIdx.x;
  int t = threadIdx.x;
  __shared__ float r1[256], r2[256];
  float su = 0.f, sv = 0.f;
  for (int d = t; d < DIMK; d += 256) {
    float z = zemb[a * DIMK + d];
    su += z * Wg[d];
    sv += z * Wg[DIMK + d];
  }
  r1[t] = su; r2[t] = sv;
  __syncthreads();
  for (int s = 128; s > 0; s >>= 1) {
    if (t < s) { r1[t] += r1[t + s]; r2[t] += r2[t + s]; }
    __syncthreads();
  }
  if (t == 0) { uv[a] = r1[0]; uv[32 + a] = r2[0]; }
}

// ---------------------------------------------------------------------------
// Kernel 4: gap score A[b] = (sum_i u[x[b,i]] + sum_j v[y[b,j]])/1024 + bg
// grid: 8 blocks x 256 threads
// ---------------------------------------------------------------------------
__global__ __launch_bounds__(256) void gap_kernel(
    const int* __restrict__ x, const int* __restrict__ y,
    const float* __restrict__ uv, const float* __restrict__ bg,
    float* __restrict__ Ag) {
  int b = blockIdx.x;
  int t = threadIdx.x;
  __shared__ float r1[256];
  float s = 0.f;
  for (int i = t; i < NSEQ; i += 256) {
    s += uv[x[b * NSEQ + i]] + uv[32 + y[b * NSEQ + i]];
  }
  r1[t] = s;
  __syncthreads();
  for (int k = 128; k > 0; k >>= 1) {
    if (t < k) r1[t] += r1[t + k];
    __syncthreads();
  }
  if (t == 0) Ag[b] = r1[0] * (1.0f / NSEQ) + bg[0];
}

// Packed anti-diagonal-major offset of diagonal s (s = i+j-2, 0..2046).
__device__ __forceinline__ int diag_off(int s) {
  return (s < NSEQ) ? ((s * (s + 1)) >> 1)
                    : (NSEQ * NSEQ - (((2 * NSEQ - 1 - s) * (2 * NSEQ - s)) >> 1));
}

// ---------------------------------------------------------------------------
// Kernel 5: forward soft-NW. One block per batch; thread t = row i = t+1.
// V stored packed diag-major (coalesced along anti-diagonal).
// ---------------------------------------------------------------------------
__global__ __launch_bounds__(1024) void nw_forward_kernel(
    const int* __restrict__ x, const int* __restrict__ y,
    const float* __restrict__ G, const float* __restrict__ Ag,
    float* __restrict__ Vg) {
  int b = blockIdx.x;
  int t = threadIdx.x;
  int i = t + 1;

  __shared__ float Gs[NSYM * 32];
  __shared__ int ys[NSEQ];
  __shared__ float B[3][NSEQ + 2];

  for (int idx = t; idx < NSYM * 32; idx += 1024) Gs[idx] = G[idx];
  ys[t] = y[b * NSEQ + t];
  int sx = x[b * NSEQ + t];  // x[b][i-1]
  float A = Ag[b];
  __syncthreads();

  size_t base = (size_t)b * (NSEQ * NSEQ);
  for (int d = 2; d <= 2 * NSEQ; ++d) {
    float* cur = B[d % 3];
    float* p1  = B[(d + 2) % 3];  // diagonal d-1
    float* p2  = B[(d + 1) % 3];  // diagonal d-2
    int j = d - i;
    if (j >= 1 && j <= NSEQ) {
      float vup   = (i == 1) ? NEGC : p1[i - 1];
      float vdiag = (i == 1) ? ((j == 1) ? 0.0f : NEGC)
                             : ((j == 1) ? NEGC : p2[i - 1]);
      float vleft = (j == 1) ? NEGC : p1[i];
      float th = Gs[sx * 32 + ys[j - 1]];
      float aa = vup + A, bb = vdiag, cc = vleft + A;
      float m = fmaxf(aa, fmaxf(bb, cc));
      float ssum = __expf(aa - m) + __expf(bb - m) + __expf(cc - m);
      float v = th + m + __logf(ssum);
      cur[i] = v;
      int s = d - 2;
      int imin = (d - NSEQ > 1) ? (d - NSEQ) : 1;
      Vg[base + diag_off(s) + (i - imin)] = v;
    }
    __syncthreads();
  }
}

// ---------------------------------------------------------------------------
// Kernel 6: backward soft-NW (gradient E = dV_final/dtheta), reverse wavefront.
// E[i,j] = [i=N,j=M] + E[i+1,j]*w_up + E[i+1,j+1]*w_diag + E[i,j+1]*w_left
// with w = exp(V[i,j] (+A) + theta(dest) - V(dest)) <= 1.
// The V anti-diagonal for the NEXT step is prefetched one step ahead with
// GLOBAL_LOAD_ASYNC_TO_LDS_B32 (ASYNCcnt) straight into the LDS rotation
// buffer, hiding the global-load latency behind the current step's compute.
// Output written transposed: out[b][j-1][i-1].
// ---------------------------------------------------------------------------
__global__ __launch_bounds__(1024) void nw_backward_kernel(
    const int* __restrict__ x, const int* __restrict__ y,
    const float* __restrict__ G, const float* __restrict__ Ag,
    const float* __restrict__ Vg, float* __restrict__ out) {
  int b = blockIdx.x;
  int t = threadIdx.x;
  int i = t + 1;

  __shared__ float Gs[NSYM * 32];
  __shared__ int ys[NSEQ];
  __shared__ float EB[3][NSEQ + 2];
  __shared__ float VB[4][NSEQ + 2];   // 4-deep: allows one-step-ahead prefetch

  for (int idx = t; idx < NSYM * 32; idx += 1024) Gs[idx] = G[idx];
  ys[t] = y[b * NSEQ + t];
  int sx_i = x[b * NSEQ + t];                           // x[i-1] (row i)
  int sx_n = (t + 1 < NSEQ) ? x[b * NSEQ + t + 1] : 0;  // x[i]   (row i+1)
  float A = Ag[b];
  __syncthreads();

  size_t base = (size_t)b * (NSEQ * NSEQ);

  // Prologue: async-prefetch V diagonal for the first step (d = 2*NSEQ).
  {
    int d = 2 * NSEQ;
    int j = d - i;
    if (j >= 1 && j <= NSEQ) {
      int s = d - 2;
      int imin = (d - NSEQ > 1) ? (d - NSEQ) : 1;
      const float* gp = Vg + base + diag_off(s) + (i - imin);
      unsigned ldsa = (unsigned)(unsigned long long)&VB[d & 3][i];
      asm volatile("global_load_async_to_lds_b32 %0, %1, off"
                   :: "v"(ldsa), "v"((unsigned long long)gp) : "memory");
    }
  }

  for (int d = 2 * NSEQ; d >= 2; --d) {
    // Wait for the async prefetch of THIS step's V diagonal (issued last step
    // or in the prologue); end-of-previous-step barrier made older diagonals
    // visible across waves.
    asm volatile("s_wait_asynccnt 0x0" ::: "memory");

    float* Ecur = EB[d % 3];
    float* E1   = EB[(d + 1) % 3];  // diagonal d+1
    float* E2   = EB[(d + 2) % 3];  // diagonal d+2
    float* Vcur = VB[d & 3];
    float* V1   = VB[(d + 1) & 3];
    float* V2   = VB[(d + 2) & 3];

    // Issue async prefetch for the NEXT step (d-1) into VB[(d+3)&3], which
    // holds diagonal d+3 whose last reader was step d+1 (barrier-separated).
    if (d > 2) {
      int jn = (d - 1) - i;
      if (jn >= 1 && jn <= NSEQ) {
        int sn = d - 3;
        int iminn = ((d - 1) - NSEQ > 1) ? ((d - 1) - NSEQ) : 1;
        const float* gp = Vg + base + diag_off(sn) + (i - iminn);
        unsigned ldsa = (unsigned)(unsigned long long)&VB[(d + 3) & 3][i];
        asm volatile("global_load_async_to_lds_b32 %0, %1, off"
                     :: "v"(ldsa), "v"((unsigned long long)gp) : "memory");
      }
    }

    int j = d - i;
    if (j >= 1 && j <= NSEQ) {
      float vij = Vcur[i];
      float e = (d == 2 * NSEQ) ? 1.0f : 0.0f;
      if (i < NSEQ) {  // up edge into (i+1, j)
        float th = Gs[sx_n * 32 + ys[j - 1]];
        e += E1[i + 1] * __expf(vij + A + th - V1[i + 1]);
      }
      if (i < NSEQ && j < NSEQ) {  // diag edge into (i+1, j+1)
        float th = Gs[sx_n * 32 + ys[j]];
        e += E2[i + 1] * __expf(vij + th - V2[i + 1]);
      }
      if (j < NSEQ) {  // left edge into (i, j+1)
        float th = Gs[sx_i * 32 + ys[j]];
        e += E1[i] * __expf(vij + A + th - V1[i]);
      }
      Ecur[i] = e;
      out[base + (size_t)(j - 1) * NSEQ + (i - 1)] = e;  // transposed output
    }
    __syncthreads();
  }
}

// ---------------------------------------------------------------------------
extern "C" void kernel_launch(void* const* d_in, const int* in_sizes, int n_in,
                              void* d_out, int out_size, void* d_ws, size_t ws_size,
                              hipStream_t stream) {
  (void)in_sizes; (void)n_in; (void)out_size; (void)ws_size;
  const int*   x    = (const int*)d_in[0];    // [8,1024]
  const int*   y    = (const int*)d_in[1];    // [8,1024]
  const float* emb  = (const float*)d_in[2];  // [22,512]
  const float* W    = (const float*)d_in[3];  // [512,512]
  const float* bias = (const float*)d_in[4];  // [512]
  const float* Wg   = (const float*)d_in[5];  // [1024,1]
  const float* bg   = (const float*)d_in[6];  // [1]
  float* out = (float*)d_out;                 // [8,1024,1024]

  float* ws   = (float*)d_ws;
  float* Vg   = ws;                                  // 8*1024*1024 floats
  float* zemb = ws + (size_t)8 * NSEQ * NSEQ;        // 32*512
  float* G    = zemb + 32 * DIMK;                    // 32*32
  float* uv   = G + 32 * 32;                         // 64
  float* Ag   = uv + 64;                             // 8

  zemb_wmma_kernel<<<64, 32, 0, stream>>>(emb, W, bias, zemb);
  gram_wmma_kernel<<<4, 32, 0, stream>>>(zemb, G);
  uv_kernel<<<NSYM, 256, 0, stream>>>(zemb, Wg, uv);
  gap_kernel<<<8, 256, 0, stream>>>(x, y, uv, bg, Ag);
  nw_forward_kernel<<<8, 1024, 0, stream>>>(x, y, G, Ag, Vg);
  nw_backward_kernel<<<8, 1024, 0, stream>>>(x, y, G, Ag, Vg, out);
}